// GINEncoder_27049704030898
// MI455X (gfx1250) — compile-verified
//
#include <hip/hip_runtime.h>

#define NN 20000
#define NE 320000
#define AIN 64
#define BIN 16
#define HID 256
#define NL 5
#define BNEPS 1e-5f

typedef __attribute__((ext_vector_type(2))) float v2f;
typedef __attribute__((ext_vector_type(8))) float v8f;

// ---------------------------------------------------------------------------
// h = x @ Wa + ba     [20000,64] x [64,256] -> [20000,256], f32 WMMA 16x16x4
// One block (8 waves) per 16-row M tile; each wave owns two 16-col N tiles.
// ---------------------------------------------------------------------------
__global__ void __launch_bounds__(256) k_init_h(const float* __restrict__ x,
                                                const float* __restrict__ Wa,
                                                const float* __restrict__ ba,
                                                float* __restrict__ h) {
  __shared__ float xs[16 * 68];  // padded stride: bank-conflict-free A frags
  const int m0 = blockIdx.x * 16;
  const int t = threadIdx.x;
  {
    int row = t >> 4;
    int col = (t & 15) * 4;
    float4 v = *reinterpret_cast<const float4*>(&x[(m0 + row) * AIN + col]);
    *reinterpret_cast<float4*>(&xs[row * 68 + col]) = v;
  }
  __syncthreads();
  const int wave = t >> 5, lane = t & 31;
  const int half = lane >> 4, ln = lane & 15;
  for (int i = 0; i < 2; ++i) {
    const int n0 = (wave * 2 + i) * 16;
    v8f acc;
    const float bias = ba[n0 + ln];
#pragma unroll
    for (int r = 0; r < 8; ++r) acc[r] = bias;
    for (int k0 = 0; k0 < AIN; k0 += 4) {
      const int kk = k0 + half * 2;
      v2f a = *reinterpret_cast<const v2f*>(&xs[ln * 68 + kk]);
      v2f b;
      b[0] = Wa[kk * HID + n0 + ln];
      b[1] = Wa[(kk + 1) * HID + n0 + ln];
      acc = __builtin_amdgcn_wmma_f32_16x16x4_f32(false, a, false, b,
                                                  (short)0, acc, false, false);
    }
#pragma unroll
    for (int r = 0; r < 8; ++r)
      h[(m0 + r + half * 8) * HID + n0 + ln] = acc[r];
  }
}

// ---------------------------------------------------------------------------
// Edge scatter: msg = relu(h[src] + (edge_attr @ We + be)); aggr[dst] += msg.
// We (16x256 = 16KB) cached in LDS. One wave per edge (grid-stride); lane
// handles channels c = lane + 32*j (fully coalesced loads & atomics).
// ---------------------------------------------------------------------------
__global__ void __launch_bounds__(256) k_scatter(const float* __restrict__ h,
                                                 const float* __restrict__ ea,
                                                 const int* __restrict__ ei,
                                                 const float* __restrict__ We,
                                                 const float* __restrict__ be,
                                                 float* __restrict__ aggr) {
  __shared__ float wes[BIN * HID];
  __shared__ float bes[HID];
  const int t = threadIdx.x;
  for (int i = t; i < BIN * HID; i += 256) wes[i] = We[i];
  bes[t] = be[t];
  __syncthreads();
  const int lane = t & 31;
  const int wg = (blockIdx.x * 256 + t) >> 5;
  const int nw = (gridDim.x * 256) >> 5;
  for (int e = wg; e < NE; e += nw) {
    const int s = ei[e];
    const int d = ei[NE + e];
    float acc[8];
#pragma unroll
    for (int j = 0; j < 8; ++j) acc[j] = bes[lane + 32 * j];
    const float4 a0 = *reinterpret_cast<const float4*>(&ea[e * BIN + 0]);
    const float4 a1 = *reinterpret_cast<const float4*>(&ea[e * BIN + 4]);
    const float4 a2 = *reinterpret_cast<const float4*>(&ea[e * BIN + 8]);
    const float4 a3 = *reinterpret_cast<const float4*>(&ea[e * BIN + 12]);
    const float av[16] = {a0.x, a0.y, a0.z, a0.w, a1.x, a1.y, a1.z, a1.w,
                          a2.x, a2.y, a2.z, a2.w, a3.x, a3.y, a3.z, a3.w};
#pragma unroll
    for (int k = 0; k < BIN; ++k) {
#pragma unroll
      for (int j = 0; j < 8; ++j)
        acc[j] = fmaf(av[k], wes[k * HID + lane + 32 * j], acc[j]);
    }
    const float* hp = &h[(size_t)s * HID];
    float* op = &aggr[(size_t)d * HID];
#pragma unroll
    for (int j = 0; j < 8; ++j) {
      float m = hp[lane + 32 * j] + acc[j];
      m = m > 0.f ? m : 0.f;
      unsafeAtomicAdd(&op[lane + 32 * j], m);
    }
  }
}

// ---------------------------------------------------------------------------
// Fused MLP: z=(1+eps)*h+aggr ; t=relu(z@W1+b1) ; u=t@W2+b2 ; + BN partials.
// One block per 16-row tile; intermediate t kept in LDS (16x260 padded).
// ---------------------------------------------------------------------------
__device__ __forceinline__ void gemm_tile(const float* __restrict__ Alds,
                                          const float* __restrict__ W,
                                          const float* __restrict__ bias,
                                          int n0, int half, int ln, v8f& acc) {
  const float b0 = bias[n0 + ln];
#pragma unroll
  for (int r = 0; r < 8; ++r) acc[r] = b0;
  for (int k0 = 0; k0 < HID; k0 += 4) {
    const int kk = k0 + half * 2;
    v2f a = *reinterpret_cast<const v2f*>(&Alds[ln * 260 + kk]);
    v2f b;
    b[0] = W[kk * HID + n0 + ln];
    b[1] = W[(kk + 1) * HID + n0 + ln];
    acc = __builtin_amdgcn_wmma_f32_16x16x4_f32(false, a, false, b, (short)0,
                                                acc, false, false);
  }
}

__global__ void __launch_bounds__(256) k_mlp(
    const float* __restrict__ h, const float* __restrict__ aggr,
    const float* __restrict__ epsArr, int layer, const float* __restrict__ W1,
    const float* __restrict__ b1, const float* __restrict__ W2,
    const float* __restrict__ b2, float* __restrict__ u,
    float* __restrict__ sums, float* __restrict__ sumsq) {
  __shared__ float zs[16 * 260];
  __shared__ float ts[16 * 260];
  const int m0 = blockIdx.x * 16;
  const int t = threadIdx.x;
  const float epsl = 1.0f + epsArr[layer];
  {
    const int row = t >> 4;
    const int col = (t & 15) * 16;
#pragma unroll
    for (int q = 0; q < 4; ++q) {
      const int c = col + q * 4;
      float4 hv = *reinterpret_cast<const float4*>(&h[(m0 + row) * HID + c]);
      float4 av = *reinterpret_cast<const float4*>(&aggr[(m0 + row) * HID + c]);
      float4 z;
      z.x = fmaf(epsl, hv.x, av.x);
      z.y = fmaf(epsl, hv.y, av.y);
      z.z = fmaf(epsl, hv.z, av.z);
      z.w = fmaf(epsl, hv.w, av.w);
      *reinterpret_cast<float4*>(&zs[row * 260 + c]) = z;
    }
  }
  __syncthreads();
  const int wave = t >> 5, lane = t & 31;
  const int half = lane >> 4, ln = lane & 15;
  // GEMM1 + relu -> ts
  for (int i = 0; i < 2; ++i) {
    const int n0 = (wave * 2 + i) * 16;
    v8f acc;
    gemm_tile(zs, W1, b1, n0, half, ln, acc);
#pragma unroll
    for (int r = 0; r < 8; ++r) {
      float v = acc[r];
      ts[(r + half * 8) * 260 + n0 + ln] = v > 0.f ? v : 0.f;
    }
  }
  __syncthreads();
  // GEMM2 -> u, plus per-channel BN partial sums
  for (int i = 0; i < 2; ++i) {
    const int n0 = (wave * 2 + i) * 16;
    v8f acc;
    gemm_tile(ts, W2, b2, n0, half, ln, acc);
    float s = 0.f, q = 0.f;
#pragma unroll
    for (int r = 0; r < 8; ++r) {
      const float v = acc[r];
      u[(size_t)(m0 + r + half * 8) * HID + n0 + ln] = v;
      s += v;
      q = fmaf(v, v, q);
    }
    unsafeAtomicAdd(&sums[n0 + ln], s);
    unsafeAtomicAdd(&sumsq[n0 + ln], q);
  }
}

// ---------------------------------------------------------------------------
// BN (population stats) + relu + residual: h += relu((u-mu)*rsqrt(var)*g+b)
// ---------------------------------------------------------------------------
__global__ void __launch_bounds__(256) k_bn(const float* __restrict__ u,
                                            const float* __restrict__ sums,
                                            const float* __restrict__ sumsq,
                                            const float* __restrict__ gamma,
                                            const float* __restrict__ beta,
                                            float* __restrict__ h) {
  const int tid = blockIdx.x * 256 + threadIdx.x;
  const int m = tid >> 6;
  if (m >= NN) return;
  const int c = (tid & 63) * 4;
  const float invN = 1.0f / (float)NN;
  float4 uv = *reinterpret_cast<const float4*>(&u[(size_t)m * HID + c]);
  float4 hv = *reinterpret_cast<const float4*>(&h[(size_t)m * HID + c]);
  float4 sv = *reinterpret_cast<const float4*>(&sums[c]);
  float4 qv = *reinterpret_cast<const float4*>(&sumsq[c]);
  float4 gv = *reinterpret_cast<const float4*>(&gamma[c]);
  float4 bv = *reinterpret_cast<const float4*>(&beta[c]);
  float uo[4] = {uv.x, uv.y, uv.z, uv.w};
  float ho[4] = {hv.x, hv.y, hv.z, hv.w};
  float so[4] = {sv.x, sv.y, sv.z, sv.w};
  float qo[4] = {qv.x, qv.y, qv.z, qv.w};
  float go[4] = {gv.x, gv.y, gv.z, gv.w};
  float bo[4] = {bv.x, bv.y, bv.z, bv.w};
#pragma unroll
  for (int j = 0; j < 4; ++j) {
    const float mu = so[j] * invN;
    const float var = fmaf(qo[j], invN, -mu * mu);
    float z = (uo[j] - mu) * __frsqrt_rn(var + BNEPS);
    z = fmaf(z, go[j], bo[j]);
    ho[j] += (z > 0.f ? z : 0.f);
  }
  float4 out = {ho[0], ho[1], ho[2], ho[3]};
  *reinterpret_cast<float4*>(&h[(size_t)m * HID + c]) = out;
}

extern "C" void kernel_launch(void* const* d_in, const int* in_sizes, int n_in,
                              void* d_out, int out_size, void* d_ws,
                              size_t ws_size, hipStream_t stream) {
  const float* x = (const float*)d_in[0];
  const float* edge_attr = (const float*)d_in[1];
  const int* edge_index = (const int*)d_in[2];
  const float* Wa = (const float*)d_in[3];
  const float* ba = (const float*)d_in[4];
  const float* We = (const float*)d_in[5];
  const float* be = (const float*)d_in[6];
  const float* eps = (const float*)d_in[7];
  const float* W1 = (const float*)d_in[8];
  const float* b1 = (const float*)d_in[9];
  const float* W2 = (const float*)d_in[10];
  const float* b2 = (const float*)d_in[11];
  const float* gamma = (const float*)d_in[12];
  const float* beta = (const float*)d_in[13];

  float* h = (float*)d_out;                 // h lives in d_out (exact size)
  float* aggr = (float*)d_ws;               // 20.48 MB
  float* u = aggr + (size_t)NN * HID;       // 20.48 MB
  float* sums = u + (size_t)NN * HID;       // 1 KB
  float* sumsq = sums + HID;                // 1 KB

  k_init_h<<<NN / 16, 256, 0, stream>>>(x, Wa, ba, h);
  for (int l = 0; l < NL; ++l) {
    hipMemsetAsync(aggr, 0, (size_t)NN * HID * sizeof(float), stream);
    hipMemsetAsync(sums, 0, 2 * HID * sizeof(float), stream);
    k_scatter<<<2048, 256, 0, stream>>>(h, edge_attr, edge_index, We, be, aggr);
    k_mlp<<<NN / 16, 256, 0, stream>>>(h, aggr, eps, l,
                                       W1 + (size_t)l * HID * HID, b1 + l * HID,
                                       W2 + (size_t)l * HID * HID, b2 + l * HID,
                                       u, sums, sumsq);
    k_bn<<<(NN * 64) / 256, 256, 0, stream>>>(u, sums, sumsq, gamma + l * HID,
                                              beta + l * HID, h);
  }
}